// MultiHeadAttentionLayer_33423435497899
// MI455X (gfx1250) — compile-verified
//
#include <hip/hip_runtime.h>

// ---------------------------------------------------------------------------
// MultiHeadAttention (cross-attention), f32, outputs (x, attention, energy).
// B=4, Lq=512, Skv=4096, H=8, dh=dv=32, IN=256, MID=256, HID=512.
//
// v_wmma_f32_16x16x4_f32 for every GEMM. The attention core is fully fused:
// energy -> masked/scaled -> softmax -> attention -> attn@V in one kernel,
// holding the per-wave 16x256 energy slice in registers so energy/attention
// are written to HBM exactly once and never re-read (~580 MB total traffic
// vs ~1.1 GB for the multi-pass version; this problem is HBM-bound).
// ---------------------------------------------------------------------------

typedef __attribute__((ext_vector_type(2))) float v2f;
typedef __attribute__((ext_vector_type(8))) float v8f;

#define B_    4
#define LQ    512
#define SKV   4096
#define NH    8
#define DH    32
#define DIN   256
#define MID   256
#define HID   512
#define NWAVE 16     // waves per fused workgroup
#define WCOLS 256    // energy columns owned per wave
#define STP   36     // LDS staging row pitch (32 + 4 pad -> conflict-free)

__device__ __forceinline__ v8f wmma4(v2f a, v2f b, v8f c) {
  // D(16x16,f32) = A(16x4,f32) x B(4x16,f32) + C
  return __builtin_amdgcn_wmma_f32_16x16x4_f32(
      false, a, false, b, (short)0, c, false, false);
}

// ---------------------------------------------------------------------------
// Generic GEMM: out[M][N] = X[M][K] * W[K][N] (+ bias). One wave per 16x64
// tile; A operand reused across 4 B tiles. grid=(M/16, N/64), block=32.
// ---------------------------------------------------------------------------
__global__ void __launch_bounds__(32)
gemm_f32_wmma(const float* __restrict__ X, const float* __restrict__ W,
              const float* __restrict__ bias, float* __restrict__ out,
              int M, int N, int K) {
  const int lane = threadIdx.x;
  const int m  = lane & 15;
  const int kh = lane >> 4;
  const int m0 = blockIdx.x * 16;
  const int n0 = blockIdx.y * 64;

  v8f acc[4] = {};
  const float* xrow = X + (size_t)(m0 + m) * K + 2 * kh;
  for (int k0 = 0; k0 < K; k0 += 4) {
    v2f a = *(const v2f*)(xrow + k0);
    const float* wr = W + (size_t)(k0 + 2 * kh) * N + n0 + m;
#pragma unroll
    for (int t = 0; t < 4; ++t) {
      v2f b;
      b.x = wr[16 * t];
      b.y = wr[N + 16 * t];
      acc[t] = wmma4(a, b, acc[t]);
    }
  }
#pragma unroll
  for (int t = 0; t < 4; ++t) {
    const int col = n0 + 16 * t + m;
    const float bv = bias ? bias[col] : 0.0f;
#pragma unroll
    for (int i = 0; i < 8; ++i)
      out[(size_t)(m0 + i + 8 * kh) * N + col] = acc[t][i] + bv;
  }
}

// ---------------------------------------------------------------------------
// Fused attention core for one (b,h,16-query) stripe.
// 16 waves; wave w owns columns [w*256, w*256+256) and keeps its 16x256
// energy slice in registers (acc[4][4] = 128 VGPRs).
// grid = (B*NH, LQ/16), block = 512.
// ---------------------------------------------------------------------------
__global__ void __launch_bounds__(512)
fused_attn(const float* __restrict__ Q, const float* __restrict__ Km,
           const float* __restrict__ V, const int* __restrict__ mask,
           float* __restrict__ energy, float* __restrict__ attn,
           float* __restrict__ xmid) {
  __shared__ float st[NWAVE][16][STP];  // wave-private transpose staging
  __shared__ float red[16][NWAVE];      // cross-wave row reductions
  __shared__ float rstat[2][16];        // [0]=row max, [1]=1/row sum
  __shared__ float xacc[16][32];        // attn@V accumulation tile

  const int tid  = threadIdx.x;
  const int lane = tid & 31;
  const int w    = tid >> 5;
  const int m    = lane & 15;
  const int kh   = lane >> 4;

  const int bh = blockIdx.x;
  const int b  = bh >> 3;
  const int h  = bh & 7;
  const int q0 = blockIdx.y * 16;
  const int sw = w * WCOLS;

  xacc[tid >> 5][tid & 31] = 0.0f;  // 512 threads cover all 512 entries

  // --- Phase 1: energy = Q K^T (WMMA), mask, scale, write, row-max ---------
  const float* qrow = Q + ((size_t)(b * LQ + q0 + m) * NH + h) * DH + 2 * kh;
  v2f qa[8];
#pragma unroll
  for (int d = 0; d < 8; ++d) qa[d] = *(const v2f*)(qrow + 4 * d);

  const float* kbase = Km + ((size_t)b * SKV * NH + h) * DH + 2 * kh;

  v8f acc[4][4] = {};
#pragma unroll
  for (int j = 0; j < 4; ++j) {
#pragma unroll
    for (int d = 0; d < 8; ++d) {
#pragma unroll
      for (int t = 0; t < 4; ++t) {
        v2f bb = *(const v2f*)(kbase +
                 (size_t)(sw + 64 * j + 16 * t + m) * (NH * DH) + 4 * d);
        acc[j][t] = wmma4(qa[d], bb, acc[j][t]);
      }
    }
  }

  const float inv_scale = 0.17677669529663687f; // 1/sqrt(32)
  float* erow = energy + ((size_t)bh * LQ + q0) * SKV;
  float vmax[8];
#pragma unroll
  for (int i = 0; i < 8; ++i) vmax[i] = -__builtin_inff();
#pragma unroll
  for (int j = 0; j < 4; ++j) {
#pragma unroll
    for (int t = 0; t < 4; ++t) {
      const int s = sw + 64 * j + 16 * t + m;
      const bool keep = mask[b * SKV + s] != 0;
#pragma unroll
      for (int i = 0; i < 8; ++i) {
        float e = keep ? acc[j][t][i] * inv_scale : -__builtin_inff();
        acc[j][t][i] = e;
        erow[(size_t)(i + 8 * kh) * SKV + s] = e;
        vmax[i] = fmaxf(vmax[i], e);
      }
    }
  }
  // reduce across the 16 lanes of each half-wave (wave32 shuffles)
#pragma unroll
  for (int i = 0; i < 8; ++i)
#pragma unroll
    for (int off = 1; off < 16; off <<= 1)
      vmax[i] = fmaxf(vmax[i], __shfl_xor(vmax[i], off, 32));
  if (m == 0) {
#pragma unroll
    for (int i = 0; i < 8; ++i) red[i + 8 * kh][w] = vmax[i];
  }
  __syncthreads();
  if (tid < 16) {
    float mx = red[tid][0];
#pragma unroll
    for (int ww = 1; ww < NWAVE; ++ww) mx = fmaxf(mx, red[tid][ww]);
    rstat[0][tid] = mx;
  }
  __syncthreads();

  // --- Phase 2: exp + row sums --------------------------------------------
  float rm[8], vsum[8];
#pragma unroll
  for (int i = 0; i < 8; ++i) { rm[i] = rstat[0][i + 8 * kh]; vsum[i] = 0.0f; }
#pragma unroll
  for (int j = 0; j < 4; ++j)
#pragma unroll
    for (int t = 0; t < 4; ++t)
#pragma unroll
      for (int i = 0; i < 8; ++i) {
        float p = __expf(acc[j][t][i] - rm[i]);  // masked cols -> exp(-inf)=0
        acc[j][t][i] = p;
        vsum[i] += p;
      }
#pragma unroll
  for (int i = 0; i < 8; ++i)
#pragma unroll
    for (int off = 1; off < 16; off <<= 1)
      vsum[i] += __shfl_xor(vsum[i], off, 32);
  if (m == 0) {
#pragma unroll
    for (int i = 0; i < 8; ++i) red[i + 8 * kh][w] = vsum[i];
  }
  __syncthreads();
  if (tid < 16) {
    float s = 0.0f;
#pragma unroll
    for (int ww = 0; ww < NWAVE; ++ww) s += red[tid][ww];
    rstat[1][tid] = 1.0f / s;
  }
  __syncthreads();

  // --- Phase 3: normalize -> attention (still in registers), write out ----
  float* arow = attn + ((size_t)bh * LQ + q0) * SKV;
  float inv[8];
#pragma unroll
  for (int i = 0; i < 8; ++i) inv[i] = rstat[1][i + 8 * kh];
#pragma unroll
  for (int j = 0; j < 4; ++j)
#pragma unroll
    for (int t = 0; t < 4; ++t) {
      const int s = sw + 64 * j + 16 * t + m;
#pragma unroll
      for (int i = 0; i < 8; ++i) {
        float p = acc[j][t][i] * inv[i];
        acc[j][t][i] = p;
        arow[(size_t)(i + 8 * kh) * SKV + s] = p;
      }
    }

  // --- Phase 4: attn @ V via wave-private LDS transpose staging -----------
  const float* vbase = V + ((size_t)b * SKV * NH + h) * DH + m;
  v8f av[2] = {};
#pragma unroll
  for (int j = 0; j < 4; ++j) {
#pragma unroll
    for (int hs = 0; hs < 2; ++hs) {
      // stage 16x32 of attention from C-layout regs into row-major LDS
#pragma unroll
      for (int tt = 0; tt < 2; ++tt)
#pragma unroll
        for (int i = 0; i < 8; ++i)
          st[w][i + 8 * kh][16 * tt + m] = acc[j][2 * hs + tt][i];
      __syncthreads();  // uniform; also orders wave-private LDS RAW
#pragma unroll
      for (int ks = 0; ks < 8; ++ks) {
        v2f a = *(const v2f*)(&st[w][m][4 * ks + 2 * kh]);  // A-operand layout
        const int sbase = sw + 64 * j + 32 * hs + 4 * ks;
        const float* vr = vbase + (size_t)(sbase + 2 * kh) * (NH * DH);
#pragma unroll
        for (int t2 = 0; t2 < 2; ++t2) {
          v2f bb;
          bb.x = vr[16 * t2];             // V[sbase+2kh  ][16t2+m]
          bb.y = vr[NH * DH + 16 * t2];   // V[sbase+2kh+1][16t2+m]
          av[t2] = wmma4(a, bb, av[t2]);
        }
      }
    }
  }

  // --- Phase 5: cross-wave reduce (ds_add_f32) and write x_mid ------------
#pragma unroll
  for (int t2 = 0; t2 < 2; ++t2)
#pragma unroll
    for (int i = 0; i < 8; ++i)
      atomicAdd(&xacc[i + 8 * kh][16 * t2 + m], av[t2][i]);
  __syncthreads();
  {
    const int r = tid >> 5, c = tid & 31;
    xmid[(size_t)(b * LQ + q0 + r) * MID + h * DH + c] = xacc[r][c];
  }
}

// ---------------------------------------------------------------------------
extern "C" void kernel_launch(void* const* d_in, const int* in_sizes, int n_in,
                              void* d_out, int out_size, void* d_ws, size_t ws_size,
                              hipStream_t stream) {
  (void)in_sizes; (void)n_in; (void)out_size; (void)ws_size;

  const float* query = (const float*)d_in[0];  // [B,LQ,DIN]
  const float* keyv  = (const float*)d_in[1];  // [B,SKV,DIN]
  const int*   mask  = (const int*)  d_in[2];  // [B,SKV]
  const float* Wq    = (const float*)d_in[3];
  const float* Wk    = (const float*)d_in[4];
  const float* Wv    = (const float*)d_in[5];
  const float* Wo    = (const float*)d_in[6];  // [MID,HID]
  const float* bo    = (const float*)d_in[7];  // [HID]

  float* out      = (float*)d_out;
  float* out_x    = out;
  float* out_attn = out_x + (size_t)B_ * LQ * HID;
  float* out_en   = out_attn + (size_t)B_ * NH * LQ * SKV;

  float* Qp = (float*)d_ws;                 // [B,LQ,NH,DH]
  float* Kp = Qp + (size_t)B_ * LQ * 256;   // [B,SKV,NH,DH]
  float* Vp = Kp + (size_t)B_ * SKV * 256;  // [B,SKV,NH,DH]
  float* Xm = Vp + (size_t)B_ * SKV * 256;  // [B,LQ,MID]

  dim3 wave(32);

  gemm_f32_wmma<<<dim3((B_ * LQ) / 16, 256 / 64), wave, 0, stream>>>(
      query, Wq, nullptr, Qp, B_ * LQ, 256, DIN);
  gemm_f32_wmma<<<dim3((B_ * SKV) / 16, 256 / 64), wave, 0, stream>>>(
      keyv, Wk, nullptr, Kp, B_ * SKV, 256, DIN);
  gemm_f32_wmma<<<dim3((B_ * SKV) / 16, 256 / 64), wave, 0, stream>>>(
      keyv, Wv, nullptr, Vp, B_ * SKV, 256, DIN);

  fused_attn<<<dim3(B_ * NH, LQ / 16), dim3(512), 0, stream>>>(
      Qp, Kp, Vp, mask, out_en, out_attn, Xm);

  gemm_f32_wmma<<<dim3((B_ * LQ) / 16, HID / 64), wave, 0, stream>>>(
      Xm, Wo, bo, out_x, B_ * LQ, HID, MID);
}